// TemporalInteraction_10926396801657
// MI455X (gfx1250) — compile-verified
//
#include <hip/hip_runtime.h>

// CDNA5 / gfx1250: wave32, V_WMMA_F32_16X16X4_F32 (D16x16 = A16x4 x B4x16 + C).
typedef __attribute__((ext_vector_type(2))) float v2f;
typedef __attribute__((ext_vector_type(8))) float v8f;

#define NDIV 8
#define FSTRIDE (256 * 56 * 56)              /* 802816 floats per frame */

constexpr int  GRID_X     = 512;
constexpr int  BLOCK      = 256;             /* 8 wave32 waves */
constexpr int  N_WAVES    = GRID_X * (BLOCK / 32);      /* 4096 waves per clip */
constexpr int  PIX_PER_IT = 32;              /* two 16-pixel WMMA tiles */
constexpr int  CHUNKS     = FSTRIDE / PIX_PER_IT;       /* 25088 */
constexpr long STEP       = (long)N_WAVES * PIX_PER_IT; /* 131072 elems = 512KB */

// out[t, p] = sum_s K[t,s] * x[s, p] + bias,  K = banded 8x8 (w0,w1,w2).
__global__ __launch_bounds__(BLOCK)
void temporal_wmma_kernel(const float* __restrict__ x,
                          const float* __restrict__ w,
                          const float* __restrict__ bias,
                          float* __restrict__ out) {
    const int lane = threadIdx.x & 31;
    const int wave = blockIdx.x * (BLOCK / 32) + (threadIdx.x >> 5);

    const float w0 = w[0], w1 = w[1], w2 = w[2];
    const float bb = bias[0];

    const int t  = lane & 15;   // A row (output frame) / B column (pixel)
    const int hi = lane >> 4;   // half-wave selects K sub-block per ISA layout

    // A = K matrix (constant): lane holds A[M=t][K = 2*hi + {0,1}] in (x,y)
    auto kv = [&](int tt, int ss) -> float {
        if (ss == tt - 1) return w0;
        if (ss == tt)     return w1;
        if (ss == tt + 1) return w2;
        return 0.0f;
    };
    v2f a1, a2;
    {
        const int   s0    = 2 * hi;
        const float valid = (t < 8) ? 1.0f : 0.0f;  // rows t=8..15 are zero
        a1.x = valid * kv(t, s0);      a1.y = valid * kv(t, s0 + 1);
        a2.x = valid * kv(t, s0 + 4);  a2.y = valid * kv(t, s0 + 5);
    }

    const size_t clip_base = (size_t)blockIdx.y * (size_t)NDIV * (size_t)FSTRIDE;

    // Lane-folded base pointers: p1 -> frame 2*hi, p2 -> frame 2*hi+4, col t.
    // All loads/prefetches become immediate offsets (FSTRIDE*4 = 3.2MB < 8.3MB).
    const float* p1 = x + clip_base + (size_t)(2 * hi) * FSTRIDE
                        + (size_t)wave * PIX_PER_IT + t;
    const float* p2 = p1 + (size_t)4 * FSTRIDE;
    float*       po = out + clip_base + (size_t)wave * PIX_PER_IT + t;

    for (int chunk = wave; chunk < CHUNKS;
         chunk += N_WAVES, p1 += STEP, p2 += STEP, po += STEP) {

        // 8-load clause, each 2x64B coalesced; every x element read exactly once
        v2f b1a, b2a, b1b, b2b;
        b1a.x = __builtin_nontemporal_load(p1);
        b1a.y = __builtin_nontemporal_load(p1 + FSTRIDE);
        b2a.x = __builtin_nontemporal_load(p2);
        b2a.y = __builtin_nontemporal_load(p2 + FSTRIDE);
        b1b.x = __builtin_nontemporal_load(p1 + 16);
        b1b.y = __builtin_nontemporal_load(p1 + FSTRIDE + 16);
        b2b.x = __builtin_nontemporal_load(p2 + 16);
        b2b.y = __builtin_nontemporal_load(p2 + FSTRIDE + 16);

        // run-ahead into GL2 for next iteration (pure immediate offsets now)
        __builtin_prefetch(p1 + STEP, 0, 0);
        __builtin_prefetch(p1 + STEP + FSTRIDE, 0, 0);
        __builtin_prefetch(p2 + STEP, 0, 0);
        __builtin_prefetch(p2 + STEP + FSTRIDE, 0, 0);

        v8f ca, cb;
        #pragma unroll
        for (int i = 0; i < 8; ++i) { ca[i] = bb; cb[i] = bb; }   // C = bias

        ca = __builtin_amdgcn_wmma_f32_16x16x4_f32(false, a1, false, b1a,
                                                   (short)0, ca, false, false);
        ca = __builtin_amdgcn_wmma_f32_16x16x4_f32(false, a2, false, b2a,
                                                   (short)0, ca, false, false);
        cb = __builtin_amdgcn_wmma_f32_16x16x4_f32(false, a1, false, b1b,
                                                   (short)0, cb, false, false);
        cb = __builtin_amdgcn_wmma_f32_16x16x4_f32(false, a2, false, b2b,
                                                   (short)0, cb, false, false);

        // D: VGPR r, lanes 0-15 = out[t=r][pixels]; lanes 16-31 = t>=8 (dropped)
        if (lane < 16) {
            #pragma unroll
            for (int r = 0; r < NDIV; ++r) {
                __builtin_nontemporal_store(ca[r], po + (size_t)r * FSTRIDE);
                __builtin_nontemporal_store(cb[r], po + (size_t)r * FSTRIDE + 16);
            }
        }
        // loop-top re-convergence restores EXEC = all ones before next WMMA
    }
}

extern "C" void kernel_launch(void* const* d_in, const int* in_sizes, int n_in,
                              void* d_out, int out_size, void* d_ws, size_t ws_size,
                              hipStream_t stream) {
    const float* x = (const float*)d_in[0];
    const float* w = (const float*)d_in[1];
    const float* b = (const float*)d_in[2];
    float* out = (float*)d_out;

    dim3 block(BLOCK, 1, 1);
    dim3 grid(GRID_X, 16, 1);   // grid.y = 16 clips; 4096 waves/clip, 6-7 chunks each
    temporal_wmma_kernel<<<grid, block, 0, stream>>>(x, w, b, out);
}